// MultiHeadAttentionQuantum_65481071403534
// MI455X (gfx1250) — compile-verified
//
#include <hip/hip_runtime.h>
#include <math.h>

// MI455X / gfx1250: wave32, WMMA 16x16 tiles.
//  K1: quantum layer collapses to cos + prefix products -> Qp (pre-scaled by
//      log2(e)/sqrt(E) so softmax uses raw v_exp_f32 = 2^x), Kp (padded K=4), V (d=2).
//  K2: flash attention per (b,h), scores computed TRANSPOSED (A=K, B=Q) via
//      V_WMMA_F32_16X16X4_F32. Scores are bounded (|s|<=3/8), so softmax needs no
//      max subtraction; each half-wave accumulates its own 8 keys/tile independently;
//      3 xor-16 shuffles merge at the end. Inner loop: 1 WMMA + 8 raw v_exp_f32 +
//      FMA chains. No cross-lane ops, no LDS, no scaling VALU, no denormal guards.
//  K3: (B*S,16) @ (16,64) projection via 4 chained V_WMMA_F32_16X16X4_F32, bias in C.

typedef float v2f __attribute__((ext_vector_type(2)));
typedef float v8f __attribute__((ext_vector_type(8)));

#define NW 8
#define SS 1024
#define EE 64

// ---------------- Kernel 1: measurement layer ----------------
// meas[0] = c1*...*c7 ; meas[w] = c0*...*cw (w=1..7), c_j = cos(x_j + theta_j)
__global__ __launch_bounds__(256)
void qmeas_kernel(const float* __restrict__ x,
                  const float* __restrict__ theta,
                  float* __restrict__ Qp,   // (64 bh, 1024 s, 4) = q * log2e/8, pad[3]=0
                  float* __restrict__ Kp,   // (64 bh, 1024 s, 4) pad[3]=0
                  float* __restrict__ V2) { // (64 bh, 1024 s, 2)
    int t = blockIdx.x * blockDim.x + threadIdx.x;  // 65536 threads
    int b   = t >> 13;
    int rem = t & 8191;
    int s   = rem >> 3;
    int h   = rem & 7;
    const float* xp = x + (((size_t)b * SS + s) * EE + h * NW);
    float c[8];
#pragma unroll
    for (int j = 0; j < 8; ++j) c[j] = cosf(xp[j] + theta[j]);
    float p1 = c[0]*c[1];
    float p2 = p1*c[2];
    float p3 = p2*c[3];
    float p4 = p3*c[4];
    float p5 = p4*c[5];
    float p6 = p5*c[6];
    float p7 = p6*c[7];
    float m0 = c[1]*c[2]*c[3]*c[4]*c[5]*c[6]*c[7];
    size_t row = (size_t)(b * NW + h) * SS + s;
    // 1/sqrt(E) * log2(e): e^(s/8) == 2^(s * 0.18033688...)
    const float sc = 0.125f * 1.4426950408889634f;
    ((float4*)Qp)[row] = make_float4(m0 * sc, p1 * sc, p2 * sc, 0.0f);
    ((float4*)Kp)[row] = make_float4(p3, p4, p5, 0.0f);
    ((float2*)V2)[row] = make_float2(p6, p7);
}

// ---------------- Kernel 2: streaming attention (transposed tiles) ----------------
// One wave per (bh, 16-query block): 64 bh * 64 blocks = 4096 waves.
// D = S^T tile: lane (n=lm) = query column, VGPR r = key row (t0 + r + 8*half).
// |score| <= 3/8 => exp never over/underflows; softmax shift-invariance lets us
// skip max tracking; exp(x) computed as native 2^x with log2e folded into Qp.
__global__ __launch_bounds__(256)
void attn_kernel(const float* __restrict__ Qp,
                 const float* __restrict__ Kp,
                 const float* __restrict__ V2,
                 float* __restrict__ out16) { // (B*S, 16) f = h*2+d
    int wave = blockIdx.x * (blockDim.x >> 5) + (threadIdx.x >> 5);
    int lane = threadIdx.x & 31;
    int half = lane >> 4;     // K-pair select for A/B operands; key-row offset +8 for D
    int lm   = lane & 15;
    int bh = wave >> 6;
    int sb = wave & 63;
    int s0 = sb << 4;
    size_t base = (size_t)bh * SS;

    // B operand (Q tile as 4x16, loop-invariant): lane n=lm, K pair = 2*half
    v2f bq = *(const v2f*)(Qp + (base + s0 + lm) * 4 + 2 * half);

    // per-lane partial sums for query s0+lm over THIS half's keys only
    float lsum = 0.f, acc0 = 0.f, acc1 = 0.f;

    for (int tb = 0; tb < 64; ++tb) {
        int t0 = tb << 4;
        // A operand (K tile, 16x4): lane m=lm (key row), K pair = 2*half
        v2f ak = *(const v2f*)(Kp + (base + t0 + lm) * 4 + 2 * half);
        v8f cz = {};
        v8f sc = __builtin_amdgcn_wmma_f32_16x16x4_f32(
            false, ak, false, bq, (short)0, cz, false, false);

        // V rows owned by this lane: t0+8*half .. +7 -> 64 contiguous bytes
        const float4* vp = (const float4*)(V2 + (base + t0 + 8 * half) * 2);
        float4 vv0 = vp[0], vv1 = vp[1], vv2 = vp[2], vv3 = vp[3];
        float v0[8] = {vv0.x, vv0.z, vv1.x, vv1.z, vv2.x, vv2.z, vv3.x, vv3.z};
        float v1[8] = {vv0.y, vv0.w, vv1.y, vv1.w, vv2.y, vv2.w, vv3.y, vv3.w};

#pragma unroll
        for (int r = 0; r < 8; ++r) {
            float p = __builtin_amdgcn_exp2f(sc[r]); // raw v_exp_f32; args bounded
            lsum += p;
            acc0 = fmaf(p, v0[r], acc0);
            acc1 = fmaf(p, v1[r], acc1);
        }
    }
    // merge the two half-wave partials (keys 0-7 vs 8-15 of every tile)
    float lt  = lsum + __shfl_xor(lsum, 16);
    float a0t = acc0 + __shfl_xor(acc0, 16);
    float a1t = acc1 + __shfl_xor(acc1, 16);

    int b = bh >> 3, h = bh & 7;
    int s = s0 + lm;
    float o = ((half == 0) ? a0t : a1t) / lt;   // half selects d component
    out16[((size_t)b * SS + s) * 16 + h * 2 + half] = o;
}

// ---------------- Kernel 3: combine projection ----------------
// (8192,16) @ W^T(16,64) + bias via 4 chained K=4 f32 WMMAs. 2048 waves.
__global__ __launch_bounds__(256)
void proj_kernel(const float* __restrict__ out16,
                 const float* __restrict__ W,     // combine_w (64,16) row-major
                 const float* __restrict__ bias,  // (64,)
                 float* __restrict__ out) {       // (8192, 64)
    int wave = blockIdx.x * (blockDim.x >> 5) + (threadIdx.x >> 5);
    int lane = threadIdx.x & 31;
    int half = lane >> 4;
    int lm   = lane & 15;
    int mtile = wave >> 2;  // 512 row tiles
    int ntile = wave & 3;   // 4 col tiles
    int m0 = mtile << 4;
    int n0 = ntile << 4;
    float bv = bias[n0 + lm];
    v8f c;
#pragma unroll
    for (int r = 0; r < 8; ++r) c[r] = bv;   // bias broadcast down the column
#pragma unroll
    for (int kc = 0; kc < 4; ++kc) {
        v2f a  = *(const v2f*)(out16 + (size_t)(m0 + lm) * 16 + kc * 4 + 2 * half);
        // B[k][n] = W[n][k]: contiguous pair along f within W row n
        v2f bb = *(const v2f*)(W + (size_t)(n0 + lm) * 16 + kc * 4 + 2 * half);
        c = __builtin_amdgcn_wmma_f32_16x16x4_f32(
            false, a, false, bb, (short)0, c, false, false);
    }
#pragma unroll
    for (int r = 0; r < 8; ++r) {
        int row = m0 + r + 8 * half;
        out[(size_t)row * 64 + n0 + lm] = c[r];
    }
}

extern "C" void kernel_launch(void* const* d_in, const int* in_sizes, int n_in,
                              void* d_out, int out_size, void* d_ws, size_t ws_size,
                              hipStream_t stream) {
    const float* x     = (const float*)d_in[0];  // (8,1024,64)
    const float* theta = (const float*)d_in[1];  // (8,)
    const float* W     = (const float*)d_in[2];  // (64,16)
    const float* bias  = (const float*)d_in[3];  // (64,)
    float* out = (float*)d_out;                  // (8,1024,64)

    char* ws = (char*)d_ws;
    float* Qp    = (float*)(ws);                              // 1 MB
    float* Kp    = (float*)(ws + (1u << 20));                 // 1 MB
    float* V2    = (float*)(ws + (2u << 20));                 // 0.5 MB
    float* out16 = (float*)(ws + (2u << 20) + (1u << 19));    // 0.5 MB

    qmeas_kernel<<<256, 256, 0, stream>>>(x, theta, Qp, Kp, V2);
    attn_kernel <<<512, 256, 0, stream>>>(Qp, Kp, V2, out16);
    proj_kernel <<<256, 256, 0, stream>>>(out16, W, bias, out);
}